// DomainAwareContrastiveLoss_9990093931190
// MI455X (gfx1250) — compile-verified
//
#include <hip/hip_runtime.h>
#include <hip/hip_bf16.h>

#define ND 2048
#define NTOT 8192
#define DEMB 256
#define KNEG 128
#define NEG_INF (-1e30f)

// workspace layout (float offsets)
#define WS_LOSS   0
#define WS_TEMPS  16
#define WS_CENTER (16 + NTOT)
#define WS_EN     (16 + NTOT + 1024)

typedef __attribute__((ext_vector_type(2))) float v2f;
typedef __attribute__((ext_vector_type(8))) float v8f;
typedef __attribute__((ext_vector_type(4))) unsigned int v4u;
typedef __attribute__((ext_vector_type(8))) int v8i;
typedef __attribute__((ext_vector_type(4))) int v4i;

#if defined(__gfx1250__) && __has_builtin(__builtin_amdgcn_tensor_load_to_lds)
#define USE_TDM 1
#else
#define USE_TDM 0
#endif

__device__ __forceinline__ float wave_sum(float x) {
#pragma unroll
  for (int o = 16; o; o >>= 1) x += __shfl_xor(x, o, 32);
  return x;
}
__device__ __forceinline__ float wave_max(float x) {
#pragma unroll
  for (int o = 16; o; o >>= 1) x = fmaxf(x, __shfl_xor(x, o, 32));
  return x;
}
// min over the 128 register-resident top-K values (4 per lane), with uniform index
__device__ __forceinline__ void min_reduce4(const float tv[4], int lane, float& m, int& mi) {
  float lm = tv[0];
  int li = 0;
#pragma unroll
  for (int t = 1; t < 4; ++t)
    if (tv[t] < lm) { lm = tv[t]; li = t; }
  li = lane * 4 + li;
#pragma unroll
  for (int o = 16; o; o >>= 1) {
    float ov = __shfl_xor(lm, o, 32);
    int   oi = __shfl_xor(li, o, 32);
    if (ov < lm || (ov == lm && oi < li)) { lm = ov; li = oi; }
  }
  m = lm; mi = li;
}

// ---------------------------------------------------------------------------
// Kernel 1: normalize rows into ws.En, compute adaptive temps, zero loss accum
// ---------------------------------------------------------------------------
__global__ __launch_bounds__(256) void dacl_prep(
    const float* __restrict__ v, const float* __restrict__ nl,
    const float* __restrict__ se, const float* __restrict__ me,
    const float* __restrict__ W1, const float* __restrict__ b1,
    const float* __restrict__ W2, const float* __restrict__ b2,
    float* __restrict__ ws) {
  const int lane = threadIdx.x & 31;
  const int wv   = threadIdx.x >> 5;
  const int r    = blockIdx.x * 8 + wv;
  if (blockIdx.x == 0 && threadIdx.x == 0) ws[WS_LOSS] = 0.0f;

  const int dom = r >> 11, loc = r & (ND - 1);
  const float* base =
      (dom == 0 ? v : dom == 1 ? nl : dom == 2 ? se : me) + (size_t)loc * DEMB;

  float e[8];
#pragma unroll
  for (int t = 0; t < 8; ++t) e[t] = base[lane + 32 * t];
  float ss = 0.0f;
#pragma unroll
  for (int t = 0; t < 8; ++t) ss += e[t] * e[t];
  ss = wave_sum(ss);
  const float inv = rsqrtf(ss);
  float* En = ws + WS_EN + (size_t)r * DEMB;
#pragma unroll
  for (int t = 0; t < 8; ++t) En[lane + 32 * t] = e[t] * inv;

  float h0 = b1[lane], h1 = b1[lane + 32];
  for (int d = 0; d < DEMB; ++d) {
    const float ev = base[d];
    h0 = fmaf(ev, W1[d * 64 + lane], h0);
    h1 = fmaf(ev, W1[d * 64 + lane + 32], h1);
  }
  h0 = fmaxf(h0, 0.0f);
  h1 = fmaxf(h1, 0.0f);
  float p = h0 * W2[lane] + h1 * W2[lane + 32];
  p = wave_sum(p);
  if (lane == 0) {
    const float t = 1.0f / (1.0f + expf(-(p + b2[0])));
    ws[WS_TEMPS + r] = 0.01f + 0.99f * t;
  }
}

// ---------------------------------------------------------------------------
// Kernel 2: per-domain centers (from UNnormalized E)
// ---------------------------------------------------------------------------
__global__ __launch_bounds__(256) void dacl_centers(
    const float* __restrict__ v, const float* __restrict__ nl,
    const float* __restrict__ se, const float* __restrict__ me,
    float* __restrict__ ws) {
  const int dom = blockIdx.x, d = threadIdx.x;
  const float* base = (dom == 0 ? v : dom == 1 ? nl : dom == 2 ? se : me);
  float s = 0.0f;
  for (int i = 0; i < ND; ++i) s += base[(size_t)i * DEMB + d];
  ws[WS_CENTER + dom * DEMB + d] = s * (1.0f / (float)ND);
}

// ---------------------------------------------------------------------------
// Kernel 3: fused cosine-similarity GEMM (f32 WMMA) + per-anchor top-128 +
//           positive pair + per-anchor temperature logsumexp -> loss sum
// ---------------------------------------------------------------------------
#define LDA 260   // A tile LDS stride (floats)
#define LDB 36    // B slice LDS stride (32 data + 4 pad; TDM pad reproduces it)
#define LDSS 132  // S tile LDS stride

#if USE_TDM
// TDM: async-load a 128-row x 32-float tile (row stride 256 floats) into LDS,
// with 4-DWORD padding after every 32 DWORDs -> LDS row stride 36 floats.
__device__ __forceinline__ void tdm_load_b_slice(const float* gsrc, float* ldsdst) {
  const unsigned long long ga = (unsigned long long)(uintptr_t)gsrc;
  v4u g0;
  g0[0] = 1u;                                        // count=1 valid descriptor
  g0[1] = (unsigned)(uintptr_t)ldsdst;               // LDS byte offset (low 32b)
  g0[2] = (unsigned)(ga & 0xffffffffu);              // global_addr[31:0]
  g0[3] = (unsigned)((ga >> 32) & 0x1ffffffu) | (2u << 30);  // addr[56:32] | type=2
  v8i g1;
  // data_size=4B, pad_enable, pad_interval=32 DWORDs (code 4), pad_amount=4 DWORDs (code 3)
  g1[0] = (2 << 16) | (1 << 20) | (4 << 22) | (3 << 25);
  g1[1] = 0;                                         // atomic barrier addr, dim0 lo16(=0)
  g1[2] = (int)(0x100000u >> 16);                    // tensor_dim0 hi16 (dim0=1M)
  g1[3] = (int)((0x100000u >> 16) | (32u << 16));    // tensor_dim1 hi16 | tile_dim0=32
  g1[4] = 128;                                       // tile_dim1=128, tile_dim2=0
  g1[5] = 256;                                       // tensor_dim0_stride lo32 = 256
  g1[6] = 0;
  g1[7] = 0;
  v4i g2 = {0, 0, 0, 0};
  v4i g3 = {0, 0, 0, 0};
#if __has_include(<hip/amd_detail/amd_gfx1250_TDM.h>)
  v8i g4 = {0, 0, 0, 0, 0, 0, 0, 0};
  __builtin_amdgcn_tensor_load_to_lds(g0, g1, g2, g3, g4, 0);
#else
  __builtin_amdgcn_tensor_load_to_lds(g0, g1, g2, g3, 0);
#endif
}
#endif

__device__ __forceinline__ void merge_anchor(const float* __restrict__ cand,
                                             float tv[4], float& m, int& mi,
                                             int lane) {
#pragma unroll
  for (int rnd = 0; rnd < 4; ++rnd) {
    float vc = cand[rnd * 32 + lane];
    for (;;) {
      const unsigned msk = (unsigned)__ballot(vc > m);
      if (!msk) break;
      const int src = __ffs(msk) - 1;
      const float val = __shfl(vc, src, 32);
      if (lane == src) vc = NEG_INF;
      const int owner = mi >> 2, slot = mi & 3;
      if (lane == owner) {
        tv[0] = (slot == 0) ? val : tv[0];
        tv[1] = (slot == 1) ? val : tv[1];
        tv[2] = (slot == 2) ? val : tv[2];
        tv[3] = (slot == 3) ? val : tv[3];
      }
      min_reduce4(tv, lane, m, mi);
    }
  }
}

__device__ __forceinline__ void finalize_anchor(int g, const float tv[4],
                                                const float* __restrict__ En,
                                                const float* __restrict__ temps,
                                                float* __restrict__ loss,
                                                int lane) {
  const int dom = g >> 11, loc = g & (ND - 1);
  const int p = (dom << 11) + ((loc + 1) & (ND - 1));
  const float* er = En + (size_t)g * DEMB;
  const float* ep = En + (size_t)p * DEMB;
  float dp = 0.0f;
#pragma unroll
  for (int t = 0; t < 8; ++t) dp = fmaf(er[lane + 32 * t], ep[lane + 32 * t], dp);
  dp = wave_sum(dp);
  const float it = 1.0f / temps[g];
  const float l0 = dp * it;
  float lt[4], mx = l0;
#pragma unroll
  for (int t = 0; t < 4; ++t) { lt[t] = tv[t] * it; mx = fmaxf(mx, lt[t]); }
  mx = wave_max(mx);
  float s = (lane == 0) ? expf(l0 - mx) : 0.0f;
#pragma unroll
  for (int t = 0; t < 4; ++t) s += expf(lt[t] - mx);
  s = wave_sum(s);
  if (lane == 0) atomicAdd(loss, logf(s) + mx - l0);
}

__global__ __launch_bounds__(256) void dacl_sim_topk(float* __restrict__ ws) {
  __shared__ float At[16 * LDA];
  __shared__ float Bs[2][128 * LDB];   // double-buffered B slice
  __shared__ float St[16 * LDSS];

  const float* En = ws + WS_EN;
  const float* temps = ws + WS_TEMPS;
  const int tid = threadIdx.x, lane = tid & 31, w = tid >> 5;
  const int rowBase = blockIdx.x * 16;
  const int rowDom = rowBase >> 11;

  // stage the 16 anchor rows once (16x256 f32)
#pragma unroll
  for (int q = 0; q < 4; ++q) {
    const int idx = tid + 256 * q;
    const int rr = idx >> 6, cc = (idx & 63) * 4;
    *(float4*)&At[rr * LDA + cc] =
        *(const float4*)&En[(size_t)(rowBase + rr) * DEMB + cc];
  }

  float tv0[4], tv1[4];
#pragma unroll
  for (int t = 0; t < 4; ++t) { tv0[t] = NEG_INF; tv1[t] = NEG_INF; }
  float m0 = NEG_INF, m1 = NEG_INF;
  int mi0 = 0, mi1 = 0;
  __syncthreads();

  const int lm = lane & 15, hi = lane >> 4;

  for (int colBase = 0; colBase < NTOT; colBase += 128) {
    if ((colBase >> 11) == rowDom) continue;  // same-domain chunk: exact skip

    const float* EnB = En + (size_t)colBase * DEMB;
    v8f acc = {0.f, 0.f, 0.f, 0.f, 0.f, 0.f, 0.f, 0.f};

#if USE_TDM
    if (w == 0) tdm_load_b_slice(EnB, &Bs[0][0]);   // prologue: slice 0 -> buf 0
    for (int kb = 0; kb < 8; ++kb) {
      __builtin_amdgcn_s_wait_tensorcnt(0);         // slice kb landed (wave 0)
      __syncthreads();                              // visible to all; prev buf free
      if (w == 0 && kb < 7)
        tdm_load_b_slice(EnB + (kb + 1) * 32, &Bs[(kb + 1) & 1][0]);  // overlap DMA
      const float* Bsl = &Bs[kb & 1][0];
#pragma unroll
      for (int kk = 0; kk < 8; ++kk) {
        const int kg = kb * 32 + kk * 4;
        v2f a = *(const v2f*)&At[lm * LDA + kg + 2 * hi];
        v2f b = *(const v2f*)&Bsl[(w * 16 + lm) * LDB + kk * 4 + 2 * hi];
        acc = __builtin_amdgcn_wmma_f32_16x16x4_f32(
            false, a, false, b, (short)0, acc, false, false);
      }
    }
#else
    for (int kb = 0; kb < 8; ++kb) {
      __syncthreads();
#pragma unroll
      for (int q = 0; q < 4; ++q) {
        const int idx = tid + 256 * q;
        const int rr = idx >> 3, cc = (idx & 7) * 4;
        *(float4*)&Bs[0][rr * LDB + cc] =
            *(const float4*)&EnB[(size_t)rr * DEMB + kb * 32 + cc];
      }
      __syncthreads();
#pragma unroll
      for (int kk = 0; kk < 8; ++kk) {
        const int kg = kb * 32 + kk * 4;
        v2f a = *(const v2f*)&At[lm * LDA + kg + 2 * hi];
        v2f b = *(const v2f*)&Bs[0][(w * 16 + lm) * LDB + kk * 4 + 2 * hi];
        acc = __builtin_amdgcn_wmma_f32_16x16x4_f32(
            false, a, false, b, (short)0, acc, false, false);
      }
    }
#endif

    // dump wave's 16x16 tile to the shared S tile
#pragma unroll
    for (int r = 0; r < 8; ++r)
      St[(r + 8 * hi) * LDSS + w * 16 + lm] = acc[r];
    __syncthreads();
    merge_anchor(&St[(2 * w) * LDSS], tv0, m0, mi0, lane);
    merge_anchor(&St[(2 * w + 1) * LDSS], tv1, m1, mi1, lane);
    __syncthreads();
  }

  finalize_anchor(rowBase + 2 * w,     tv0, En, temps, ws + WS_LOSS, lane);
  finalize_anchor(rowBase + 2 * w + 1, tv1, En, temps, ws + WS_LOSS, lane);
}

// ---------------------------------------------------------------------------
// Kernel 4: domain-alignment reg + combine
// ---------------------------------------------------------------------------
__global__ __launch_bounds__(256) void dacl_final(
    const float* __restrict__ dw, float* __restrict__ ws,
    float* __restrict__ out) {
  __shared__ float red[256];
  const int t = threadIdx.x;
  float reg = 0.0f;
  const float* C = ws + WS_CENTER;
  for (int i = 0; i < 4; ++i) {
    for (int j = i + 1; j < 4; ++j) {
      const float diff = C[i * DEMB + t] - C[j * DEMB + t];
      red[t] = diff * diff;
      __syncthreads();
      for (int o = 128; o; o >>= 1) {
        if (t < o) red[t] += red[t + o];
        __syncthreads();
      }
      if (t == 0) reg += dw[i * 4 + j] * sqrtf(red[0]);
      __syncthreads();
    }
  }
  if (t == 0)
    out[0] = ws[WS_LOSS] * (1.0f / (float)NTOT) + 0.5f * (reg * (1.0f / 6.0f));
}

extern "C" void kernel_launch(void* const* d_in, const int* in_sizes, int n_in,
                              void* d_out, int out_size, void* d_ws,
                              size_t ws_size, hipStream_t stream) {
  const float* v  = (const float*)d_in[0];
  const float* nl = (const float*)d_in[1];
  const float* se = (const float*)d_in[2];
  const float* me = (const float*)d_in[3];
  const float* W1 = (const float*)d_in[4];
  const float* b1 = (const float*)d_in[5];
  const float* W2 = (const float*)d_in[6];
  const float* b2 = (const float*)d_in[7];
  const float* dw = (const float*)d_in[9];
  float* ws  = (float*)d_ws;
  float* out = (float*)d_out;

  dacl_prep<<<NTOT / 8, 256, 0, stream>>>(v, nl, se, me, W1, b1, W2, b2, ws);
  dacl_centers<<<4, 256, 0, stream>>>(v, nl, se, me, ws);
  dacl_sim_topk<<<NTOT / 16, 256, 0, stream>>>(ws);
  dacl_final<<<1, 256, 0, stream>>>(dw, ws, out);
}